// LightGCN_26388279066879
// MI455X (gfx1250) — compile-verified
//
#include <hip/hip_runtime.h>

#define EMB_D 64
#define EMB_DV4 16

typedef __attribute__((ext_vector_type(2))) float v2f;
typedef __attribute__((ext_vector_type(8))) float v8f;

// Guaranteed-hardware fp32 atomic add (no-return form, STOREcnt tracked).
// Uses one 64-bit base VGPR pair + immediate offset per element.
__device__ __forceinline__ void atom_add_f32_off0(float* base, float v) {
  asm volatile("global_atomic_add_f32 %0, %1, off" : : "v"(base), "v"(v) : "memory");
}
__device__ __forceinline__ void atom_add_f32_off4(float* base, float v) {
  asm volatile("global_atomic_add_f32 %0, %1, off offset:4" : : "v"(base), "v"(v) : "memory");
}
__device__ __forceinline__ void atom_add_f32_off8(float* base, float v) {
  asm volatile("global_atomic_add_f32 %0, %1, off offset:8" : : "v"(base), "v"(v) : "memory");
}
__device__ __forceinline__ void atom_add_f32_off12(float* base, float v) {
  asm volatile("global_atomic_add_f32 %0, %1, off offset:12" : : "v"(base), "v"(v) : "memory");
}

// ---------------------------------------------------------------------------
// Kernel 1: x0 = concat(user_table, item_table[1:]) ; acc = x0 ; x_next = 0
// ---------------------------------------------------------------------------
__global__ void lgcn_build_x0(const float* __restrict__ user_table,
                              const float* __restrict__ item_table,
                              float* __restrict__ x,
                              float* __restrict__ acc,
                              float* __restrict__ xnext,
                              int usernum, long long total4) {
  long long i = (long long)blockIdx.x * blockDim.x + threadIdx.x;
  if (i >= total4) return;
  int node = (int)(i >> 4);
  int q    = (int)(i & 15);
  float4 v;
  if (node <= usernum) {
    v = ((const float4*)user_table)[node * EMB_DV4 + q];
  } else {
    int r = node - usernum;  // item row 1..itemnum
    v = ((const float4*)item_table)[r * EMB_DV4 + q];
  }
  ((float4*)x)[i]     = v;
  ((float4*)acc)[i]   = v;
  ((float4*)xnext)[i] = make_float4(0.f, 0.f, 0.f, 0.f);
}

// ---------------------------------------------------------------------------
// Kernel 2: SpMM scatter: y[rows[e]] += vals[e] * x[cols[e]]
// 16 threads per edge, one float4 (4 dims) per thread, 4 HW fp32 atomics.
// Edge lists stream once (non-temporal); x gathers are L2-resident (38 MB).
// ---------------------------------------------------------------------------
__global__ void lgcn_spmm_scatter(const int* __restrict__ rows,
                                  const int* __restrict__ cols,
                                  const float* __restrict__ vals,
                                  const float* __restrict__ x,
                                  float* __restrict__ y,
                                  int nnz) {
  long long t = (long long)blockIdx.x * blockDim.x + threadIdx.x;
  if (t >= (long long)nnz * EMB_DV4) return;
  int e = (int)(t >> 4);
  int q = (int)(t & 15);
  int   r = __builtin_nontemporal_load(rows + e);
  int   c = __builtin_nontemporal_load(cols + e);
  float v = __builtin_nontemporal_load(vals + e);
  float4 xv = ((const float4*)x)[c * EMB_DV4 + q];
  float* yp = y + ((long long)r * EMB_D + q * 4);
  atom_add_f32_off0 (yp, v * xv.x);
  atom_add_f32_off4 (yp, v * xv.y);
  atom_add_f32_off8 (yp, v * xv.z);
  atom_add_f32_off12(yp, v * xv.w);
}

// ---------------------------------------------------------------------------
// Kernel 3: acc += y ; optionally zero z (dead previous-layer buffer, which
// becomes next layer's scatter destination).
// ---------------------------------------------------------------------------
__global__ void lgcn_acc_add_zero(float* __restrict__ acc,
                                  const float* __restrict__ y,
                                  float* __restrict__ z,
                                  int do_zero, long long total4) {
  long long i = (long long)blockIdx.x * blockDim.x + threadIdx.x;
  if (i >= total4) return;
  float4 a = ((const float4*)acc)[i];
  float4 b = ((const float4*)y)[i];
  a.x += b.x; a.y += b.y; a.z += b.z; a.w += b.w;
  ((float4*)acc)[i] = a;
  if (do_zero) ((float4*)z)[i] = make_float4(0.f, 0.f, 0.f, 0.f);
}

// ---------------------------------------------------------------------------
// Kernel 4: readout. One wave per 16-batch tile:
//   pos[b] = (acc[u] . acc[p]) / 16 ,  neg[b] = (acc[u] . acc[n]) / 16
// via V_WMMA_F32_16X16X4_F32 accumulated over K=64, diagonal extracted.
// ---------------------------------------------------------------------------
__global__ void lgcn_logits(const float* __restrict__ acc,
                            const int* __restrict__ user_ids,
                            const int* __restrict__ pos_seqs,
                            const int* __restrict__ neg_seqs,
                            float* __restrict__ out,
                            int usernum, int itemnum, int batch) {
  int tile = blockIdx.x;
  int lane = threadIdx.x;        // 0..31, one wave per block
  int ntiles = batch >> 4;
  if (tile >= ntiles) return;
  int n = lane & 15;
  int h = lane >> 4;
  int b = tile * 16 + n;

  int uid = user_ids[b]; uid = uid < 0 ? 0 : (uid > usernum ? usernum : uid);
  int pid = pos_seqs[b]; pid = pid < 1 ? 1 : (pid > itemnum ? itemnum : pid);
  int nid = neg_seqs[b]; nid = nid < 1 ? 1 : (nid > itemnum ? itemnum : nid);

  const float* urow = acc + (long long)uid * EMB_D;
  const float* prow = acc + (long long)(usernum + pid) * EMB_D;
  const float* nrow = acc + (long long)(usernum + nid) * EMB_D;

#if defined(__gfx1250__) && __has_builtin(__builtin_amdgcn_wmma_f32_16x16x4_f32)
  v8f cpos = {};
  v8f cneg = {};
  #pragma unroll
  for (int k0 = 0; k0 < EMB_D; k0 += 4) {
    v2f a, bp, bn;
    a.x  = urow[k0 + 2 * h]; a.y  = urow[k0 + 2 * h + 1];
    bp.x = prow[k0 + 2 * h]; bp.y = prow[k0 + 2 * h + 1];
    bn.x = nrow[k0 + 2 * h]; bn.y = nrow[k0 + 2 * h + 1];
    cpos = __builtin_amdgcn_wmma_f32_16x16x4_f32(false, a, false, bp,
                                                 (short)0, cpos, false, false);
    cneg = __builtin_amdgcn_wmma_f32_16x16x4_f32(false, a, false, bn,
                                                 (short)0, cneg, false, false);
  }
  // D layout: vgpr v holds (M = v + 8h, N = n). Diagonal element where M == N.
  #pragma unroll
  for (int v = 0; v < 8; v++) {
    int M = v + (h ? 8 : 0);
    if (M == n) {
      out[b]         = cpos[v] * 0.0625f;
      out[batch + b] = cneg[v] * 0.0625f;
    }
  }
#else
  // Fallback: half-row dot per lane + cross-half shuffle reduce.
  float sp = 0.f, sn = 0.f;
  #pragma unroll
  for (int k = 0; k < 32; k++) {
    float u = urow[h * 32 + k];
    sp += u * prow[h * 32 + k];
    sn += u * nrow[h * 32 + k];
  }
  sp += __shfl_xor(sp, 16, 32);
  sn += __shfl_xor(sn, 16, 32);
  if (h == 0) {
    out[b]         = sp * 0.0625f;
    out[batch + b] = sn * 0.0625f;
  }
#endif
}

// ---------------------------------------------------------------------------
// Host-side orchestration
// ---------------------------------------------------------------------------
extern "C" void kernel_launch(void* const* d_in, const int* in_sizes, int n_in,
                              void* d_out, int out_size, void* d_ws, size_t ws_size,
                              hipStream_t stream) {
  const float* user_table = (const float*)d_in[0];
  const float* item_table = (const float*)d_in[1];
  const float* vals       = (const float*)d_in[2];
  const int*   rows       = (const int*)d_in[3];
  const int*   cols       = (const int*)d_in[4];
  const int*   user_ids   = (const int*)d_in[5];
  const int*   pos_seqs   = (const int*)d_in[6];
  const int*   neg_seqs   = (const int*)d_in[7];

  const int usernum = in_sizes[0] / EMB_D - 1;
  const int itemnum = in_sizes[1] / EMB_D - 1;
  const int n_nodes = usernum + itemnum + 1;
  const int nnz     = in_sizes[2];
  const int batch   = in_sizes[5];

  const size_t npf = (size_t)n_nodes * EMB_D;  // floats per table
  float* x_cur  = (float*)d_ws;
  float* x_next = x_cur + npf;
  float* acc    = x_next + npf;

  const int tb = 256;
  const long long total4 = (long long)n_nodes * EMB_DV4;
  const int node_blocks = (int)((total4 + tb - 1) / tb);

  lgcn_build_x0<<<node_blocks, tb, 0, stream>>>(user_table, item_table,
                                                x_cur, acc, x_next,
                                                usernum, total4);

  const long long scat_threads = (long long)nnz * EMB_DV4;
  const int scat_blocks = (int)((scat_threads + tb - 1) / tb);

  for (int layer = 0; layer < 3; layer++) {
    lgcn_spmm_scatter<<<scat_blocks, tb, 0, stream>>>(rows, cols, vals,
                                                      x_cur, x_next, nnz);
    // acc += x_next; zero the now-dead x_cur (next layer's destination).
    lgcn_acc_add_zero<<<node_blocks, tb, 0, stream>>>(acc, x_next, x_cur,
                                                      layer < 2 ? 1 : 0,
                                                      total4);
    float* tmp = x_cur; x_cur = x_next; x_next = tmp;
  }

  lgcn_logits<<<batch / 16, 32, 0, stream>>>(acc, user_ids, pos_seqs, neg_seqs,
                                             (float*)d_out, usernum, itemnum,
                                             batch);
}